// Correlation2D_10969346474477
// MI455X (gfx1250) — compile-verified
//
#include <hip/hip_runtime.h>
#include <math.h>

#define BB 4
#define HH 256
#define WW 256
#define CC 16
#define NTR 9

typedef float v2f __attribute__((ext_vector_type(2)));
typedef float v8f __attribute__((ext_vector_type(8)));

// LDS tile is 16x16 with row stride 17 (bank-conflict padding). Natural index i -> cell.
__device__ __forceinline__ int SIDX1(int i) { return (i >> 4) * 17 + (i & 15); }

// ---------------------------------------------------------------------------
// One 16x16x16 COMPLEX matmul  D = F16 * M, where F16[k][n] = exp(sign*2pi*i*k*n/16)
// M is read from LDS cell [K*17 + N]. Result D in WMMA C-layout:
//   component v, lane l:  D[ M = v + 8*(l>>4) ][ N = l&15 ]
// Uses 16 x v_wmma_f32_16x16x4_f32 (4 K-chunks x 4 real matmuls).
// ---------------------------------------------------------------------------
__device__ __forceinline__ void cmm16(const float* __restrict__ sR,
                                      const float* __restrict__ sI,
                                      int lane, float sign, v8f& Dr, v8f& Di) {
  const int n = lane & 15;
  const int g = lane >> 4;
  Dr = (v8f){};
  Di = (v8f){};
#pragma unroll
  for (int j = 0; j < 4; ++j) {
    v2f Ar, Ai, AiN, Br, Bi;
#pragma unroll
    for (int r = 0; r < 2; ++r) {
      const int kk = 4 * j + r + 2 * g;        // K index (0..15)
      const int ph = (n * kk) & 15;            // phase mod 16
      float si_, co_;
      __sincosf(sign * (6.28318530717958647692f / 16.0f) * (float)ph, &si_, &co_);
      Ar[r] = co_;
      Ai[r] = si_;
      AiN[r] = -si_;
      Br[r] = sR[kk * 17 + n];
      Bi[r] = sI[kk * 17 + n];
    }
    // Dr += Fr*Mr - Fi*Mi ; Di += Fr*Mi + Fi*Mr   (F32 WMMA has no A/B neg -> pre-negated Ai)
    Dr = __builtin_amdgcn_wmma_f32_16x16x4_f32(false, Ar, false, Br, (short)0, Dr, false, false);
    Dr = __builtin_amdgcn_wmma_f32_16x16x4_f32(false, AiN, false, Bi, (short)0, Dr, false, false);
    Di = __builtin_amdgcn_wmma_f32_16x16x4_f32(false, Ar, false, Bi, (short)0, Di, false, false);
    Di = __builtin_amdgcn_wmma_f32_16x16x4_f32(false, Ai, false, Br, (short)0, Di, false, false);
  }
}

// ---------------------------------------------------------------------------
// In-place 256-point complex FFT of one wave's LDS buffer (natural order in/out).
// Radix-16^2 Cooley-Tukey: stage1 matmul, twiddle, transpose-store, stage2 matmul.
// sign = -1 forward (numpy convention), +1 inverse (unnormalized).
// ---------------------------------------------------------------------------
__device__ __forceinline__ void cfft256(float* sR, float* sI, int lane, float sign) {
  const int n = lane & 15;
  const int g = lane >> 4;

  v8f Yr, Yi;
  cmm16(sR, sI, lane, sign, Yr, Yi);        // Y[k1][n2] = sum_n1 F[k1][n1] M[n1][n2]

  // twiddle W256^(k1*n2) and store transposed: cell [n2*17 + k1]
#pragma unroll
  for (int v = 0; v < 8; ++v) {
    const int k1 = v + 8 * g;
    const int ph = (k1 * n) & 255;
    float si_, co_;
    __sincosf(sign * (6.28318530717958647692f / 256.0f) * (float)ph, &si_, &co_);
    const float tr = Yr[v] * co_ - Yi[v] * si_;
    const float ti = Yr[v] * si_ + Yi[v] * co_;
    sR[n * 17 + k1] = tr;
    sI[n * 17 + k1] = ti;
  }
  asm volatile("s_wait_dscnt 0" ::: "memory");
  __builtin_amdgcn_wave_barrier();

  v8f Zr, Zi;
  cmm16(sR, sI, lane, sign, Zr, Zi);        // Z[k2][k1] = sum_n2 F[k2][n2] Y'[k1][n2]

  // X[16*k2 + k1] = Z[k2][k1] -> natural-order cell [k2*17 + k1]
#pragma unroll
  for (int v = 0; v < 8; ++v) {
    sR[(v + 8 * g) * 17 + n] = Zr[v];
    sI[(v + 8 * g) * 17 + n] = Zi[v];
  }
  asm volatile("s_wait_dscnt 0" ::: "memory");
  __builtin_amdgcn_wave_barrier();
}

// ---------------------------------------------------------------------------
// Nearest-neighbor projective warp sample (matches jnp.round = round-to-even).
// ---------------------------------------------------------------------------
__device__ __forceinline__ float warp_sample(const float* __restrict__ inp,
                                             float a0, float a1, float a2, float a3,
                                             float a4, float a5, float a6, float a7,
                                             int b, int c, int h, int w) {
  const float xx = (float)w, yy = (float)h;
  const float k = a6 * xx + a7 * yy + 1.0f;
  const float xi = rintf((a0 * xx + a1 * yy + a2) / k);
  const float yi = rintf((a3 * xx + a4 * yy + a5) / k);
  const bool valid = (xi >= 0.0f) && (xi <= (float)(WW - 1)) &&
                     (yi >= 0.0f) && (yi <= (float)(HH - 1));
  const int xic = valid ? (int)xi : 0;
  const int yic = valid ? (int)yi : 0;
  const float v = inp[(((size_t)b * HH + yic) * WW + xic) * CC + c];
  return valid ? v : 0.0f;
}

// ---------------------------------------------------------------------------
// Per-channel L2 norms: slot 0 -> nx[b,c] (plain x), slots 1..9 -> ny[nt,b,c] (warped).
// ---------------------------------------------------------------------------
__global__ void __launch_bounds__(256)
k_norms(const float* __restrict__ inp, const float* __restrict__ tf,
        float* __restrict__ nx, float* __restrict__ ny) {
  const int slot = blockIdx.x / (BB * CC);
  const int rem = blockIdx.x % (BB * CC);
  const int b = rem / CC;
  const int c = rem % CC;
  __shared__ float red[256];
  float acc = 0.0f;
  if (slot == 0) {
    for (int i = threadIdx.x; i < HH * WW; i += 256) {
      const float v = inp[((size_t)b * HH * WW + i) * CC + c];
      acc += v * v;
    }
  } else {
    const float* t = tf + (size_t)(slot - 1) * 8;
    const float a0 = t[0], a1 = t[1], a2 = t[2], a3 = t[3];
    const float a4 = t[4], a5 = t[5], a6 = t[6], a7 = t[7];
    for (int i = threadIdx.x; i < HH * WW; i += 256) {
      const int h = i >> 8, w = i & 255;
      const float v = warp_sample(inp, a0, a1, a2, a3, a4, a5, a6, a7, b, c, h, w);
      acc += v * v;
    }
  }
  red[threadIdx.x] = acc;
  __syncthreads();
  for (int s = 128; s > 0; s >>= 1) {
    if ((int)threadIdx.x < s) red[threadIdx.x] += red[threadIdx.x + s];
    __syncthreads();
  }
  if (threadIdx.x == 0) {
    const float v = sqrtf(red[0]);
    if (slot == 0) nx[b * CC + c] = v;
    else ny[((slot - 1) * BB + b) * CC + c] = v;
  }
}

// ---------------------------------------------------------------------------
// Row FFTs. One wave per row. mode 0: rows of x (channel-last input, imag=0).
// mode 1: rows of warped y (sampled on the fly). mode 2: complex planes src.
// ---------------------------------------------------------------------------
__global__ void __launch_bounds__(256)
k_fft_rows(const float* __restrict__ inp, const float* __restrict__ tf,
           const float* __restrict__ srcR, const float* __restrict__ srcI,
           float* __restrict__ dstR, float* __restrict__ dstI,
           int mode, int nt, float sign) {
  __shared__ float smem[8 * 544];
  const int lane = threadIdx.x & 31;
  const int wv = threadIdx.x >> 5;
  float* sR = &smem[wv * 544];
  float* sI = sR + 272;
  const int gid = blockIdx.x * 8 + wv;
  const int img = gid >> 8;
  const int row = gid & 255;

  if (mode == 0) {
    const int b = img >> 4, c = img & 15;
    for (int i = lane; i < 256; i += 32) {
      sR[SIDX1(i)] = inp[(((size_t)b * HH + row) * WW + i) * CC + c];
      sI[SIDX1(i)] = 0.0f;
    }
  } else if (mode == 1) {
    const int b = img >> 4, c = img & 15;
    const float* t = tf + (size_t)nt * 8;
    const float a0 = t[0], a1 = t[1], a2 = t[2], a3 = t[3];
    const float a4 = t[4], a5 = t[5], a6 = t[6], a7 = t[7];
    for (int i = lane; i < 256; i += 32) {
      sR[SIDX1(i)] = warp_sample(inp, a0, a1, a2, a3, a4, a5, a6, a7, b, c, row, i);
      sI[SIDX1(i)] = 0.0f;
    }
  } else {
    const size_t base = ((size_t)img * HH + row) * WW;
    for (int i = lane; i < 256; i += 32) {
      sR[SIDX1(i)] = srcR[base + i];
      sI[SIDX1(i)] = srcI[base + i];
    }
  }
  asm volatile("s_wait_dscnt 0" ::: "memory");
  __builtin_amdgcn_wave_barrier();

  cfft256(sR, sI, lane, sign);

  const size_t obase = ((size_t)img * HH + row) * WW;
  for (int i = lane; i < 256; i += 32) {
    dstR[obase + i] = sR[SIDX1(i)];
    dstI[obase + i] = sI[SIDX1(i)];
  }
}

// ---------------------------------------------------------------------------
// Column FFTs of complex planes. One wave per column.
// ---------------------------------------------------------------------------
__global__ void __launch_bounds__(256)
k_fft_cols(const float* __restrict__ srcR, const float* __restrict__ srcI,
           float* __restrict__ dstR, float* __restrict__ dstI, float sign) {
  __shared__ float smem[8 * 544];
  const int lane = threadIdx.x & 31;
  const int wv = threadIdx.x >> 5;
  float* sR = &smem[wv * 544];
  float* sI = sR + 272;
  const int gid = blockIdx.x * 8 + wv;
  const int img = gid >> 8;
  const int col = gid & 255;

  for (int i = lane; i < 256; i += 32) {
    const size_t a = ((size_t)img * HH + i) * WW + col;
    sR[SIDX1(i)] = srcR[a];
    sI[SIDX1(i)] = srcI[a];
  }
  asm volatile("s_wait_dscnt 0" ::: "memory");
  __builtin_amdgcn_wave_barrier();

  cfft256(sR, sI, lane, sign);

  for (int i = lane; i < 256; i += 32) {
    const size_t a = ((size_t)img * HH + i) * WW + col;
    dstR[a] = sR[SIDX1(i)];
    dstI[a] = sI[SIDX1(i)];
  }
}

// ---------------------------------------------------------------------------
// Column FFT of warped-y row-FFT result, fused with frequency-domain
// accumulation S[nt,b] = sum_c  Fx[b,c] * conj(Fy[nt,b,c]) / (nx*ny + 1e-12).
// One wave per (b, col); channel loop in registers -> each S element written once.
// ---------------------------------------------------------------------------
__global__ void __launch_bounds__(256)
k_fftcol_accum(const float* __restrict__ t1R, const float* __restrict__ t1I,
               const float* __restrict__ FxR, const float* __restrict__ FxI,
               const float* __restrict__ nx, const float* __restrict__ ny,
               float* __restrict__ SRo, float* __restrict__ SIo, int nt) {
  __shared__ float smem[8 * 544];
  const int lane = threadIdx.x & 31;
  const int wv = threadIdx.x >> 5;
  float* sR = &smem[wv * 544];
  float* sI = sR + 272;
  const int gid = blockIdx.x * 8 + wv;
  const int b = gid >> 8;
  const int col = gid & 255;

  float accR[8], accI[8];
#pragma unroll
  for (int t = 0; t < 8; ++t) { accR[t] = 0.0f; accI[t] = 0.0f; }

  for (int c = 0; c < CC; ++c) {
    const int img = b * CC + c;
    for (int i = lane; i < 256; i += 32) {
      const size_t a = ((size_t)img * HH + i) * WW + col;
      sR[SIDX1(i)] = t1R[a];
      sI[SIDX1(i)] = t1I[a];
    }
    asm volatile("s_wait_dscnt 0" ::: "memory");
    __builtin_amdgcn_wave_barrier();

    cfft256(sR, sI, lane, -1.0f);

    const float sc = 1.0f / (nx[img] * ny[(nt * BB + b) * CC + c] + 1e-12f);
#pragma unroll
    for (int t = 0; t < 8; ++t) {
      const int k = lane + 32 * t;
      const float fyR = sR[SIDX1(k)];
      const float fyI = sI[SIDX1(k)];
      const size_t a = ((size_t)img * HH + k) * WW + col;
      const float fxR = FxR[a];
      const float fxI = FxI[a];
      accR[t] += sc * (fxR * fyR + fxI * fyI);   // Re(Fx * conj(Fy))
      accI[t] += sc * (fxI * fyR - fxR * fyI);   // Im(Fx * conj(Fy))
    }
    __builtin_amdgcn_wave_barrier();
  }

#pragma unroll
  for (int t = 0; t < 8; ++t) {
    const int k = lane + 32 * t;
    const size_t a = (((size_t)nt * BB + b) * HH + k) * WW + col;
    SRo[a] = accR[t];
    SIo[a] = accI[t];
  }
}

// ---------------------------------------------------------------------------
// Inverse column FFT of the row-inverse-FFT'd S; write real part, scaled by
// 1/(H*W*C), scattered into (B,H,W,NT) output. One wave per (nt*B+b, col).
// ---------------------------------------------------------------------------
__global__ void __launch_bounds__(256)
k_ifft_cols_out(const float* __restrict__ t1R, const float* __restrict__ t1I,
                float* __restrict__ out) {
  __shared__ float smem[8 * 544];
  const int lane = threadIdx.x & 31;
  const int wv = threadIdx.x >> 5;
  float* sR = &smem[wv * 544];
  float* sI = sR + 272;
  const int gid = blockIdx.x * 8 + wv;
  const int m = gid >> 8;            // m = nt*BB + b
  const int col = gid & 255;
  const int nt = m / BB;
  const int b = m % BB;

  for (int i = lane; i < 256; i += 32) {
    const size_t a = ((size_t)m * HH + i) * WW + col;
    sR[SIDX1(i)] = t1R[a];
    sI[SIDX1(i)] = t1I[a];
  }
  asm volatile("s_wait_dscnt 0" ::: "memory");
  __builtin_amdgcn_wave_barrier();

  cfft256(sR, sI, lane, +1.0f);

  const float scale = 1.0f / ((float)HH * (float)WW * (float)CC);
#pragma unroll
  for (int t = 0; t < 8; ++t) {
    const int h = lane + 32 * t;
    out[(((size_t)b * HH + h) * WW + col) * NTR + nt] = sR[SIDX1(h)] * scale;
  }
}

// ---------------------------------------------------------------------------
extern "C" void kernel_launch(void* const* d_in, const int* in_sizes, int n_in,
                              void* d_out, int out_size, void* d_ws, size_t ws_size,
                              hipStream_t stream) {
  const float* inp = (const float*)d_in[0];   // (B,H,W,C) f32
  const float* tf = (const float*)d_in[1];    // (NT,8) f32
  float* out = (float*)d_out;                 // (B,H,W,NT) f32
  float* ws = (float*)d_ws;

  const size_t IMG = (size_t)HH * WW;         // 65536
  float* FxR = ws;                            // 64 images
  float* FxI = FxR + 64 * IMG;
  float* T1R = FxI + 64 * IMG;                // reused: x row-fft, y row-fft, S row-ifft
  float* T1I = T1R + 64 * IMG;
  float* SR  = T1I + 64 * IMG;                // 36 images (NT*B)
  float* SI  = SR + 36 * IMG;
  float* nxp = SI + 36 * IMG;                 // B*C
  float* nyp = nxp + 64;                      // NT*B*C

  // per-channel norms (x and all 9 warps)
  k_norms<<<(NTR + 1) * BB * CC, 256, 0, stream>>>(inp, tf, nxp, nyp);

  // Fx = fft2(x) for all (b,c)
  k_fft_rows<<<BB * CC * HH / 8, 256, 0, stream>>>(inp, tf, nullptr, nullptr,
                                                   T1R, T1I, 0, 0, -1.0f);
  k_fft_cols<<<BB * CC * WW / 8, 256, 0, stream>>>(T1R, T1I, FxR, FxI, -1.0f);

  // per transform: fft2(warp(x)) fused with normalized cross-spectrum accumulation
  for (int nt = 0; nt < NTR; ++nt) {
    k_fft_rows<<<BB * CC * HH / 8, 256, 0, stream>>>(inp, tf, nullptr, nullptr,
                                                     T1R, T1I, 1, nt, -1.0f);
    k_fftcol_accum<<<BB * WW / 8, 256, 0, stream>>>(T1R, T1I, FxR, FxI,
                                                    nxp, nyp, SR, SI, nt);
  }

  // corr = real(ifft2(S)) / (H*W*C), scattered to (B,H,W,NT)
  k_fft_rows<<<NTR * BB * HH / 8, 256, 0, stream>>>(inp, tf, SR, SI,
                                                    T1R, T1I, 2, 0, +1.0f);
  k_ifft_cols_out<<<NTR * BB * WW / 8, 256, 0, stream>>>(T1R, T1I, out);
}